// SurrogateAUCLossU1_4406636445811
// MI455X (gfx1250) — compile-verified
//
#include <hip/hip_runtime.h>

// Surrogate AUC loss on MI455X (gfx1250, wave32).
//
// num = sum_{i,j} pos_i * neg_j * max(0, 1 - yp_i + yp_j),  out = num/(n_pos*n_neg)
//
// Key identity (pos, neg in {0,1}):
//   pos_i*neg_j*max(0, a_i + b_j) == max(0, pos_i*neg_j*(a_i + b_j))
// with a_i = 1 - yp_i, b_j = yp_j. The rank-2 bilinear part is computed with
// V_WMMA_F32_16X16X4_F32 as two chained K=1 products (only the K=0 slot of the
// A/B fragments is populated -> layout-robust; zeros elsewhere contribute 0):
//   D = (pos*a) x (neg)  then  D += (pos) x (neg*b)
// ReLU of the 16x16 f32 C/D fragment is a single v_med3_f32 per element
// (med3(x, 0, FLT_MAX) == max(0, x) for finite x; WMMA output is never NaN),
// avoiding the 2-op canonicalize+max sequence fmaxf generates.
//
// All reductions use fixed-order shuffle/LDS trees (no float atomics) so the
// result is bit-deterministic across graph replays.

typedef __attribute__((ext_vector_type(2))) float v2f; // 16x4 f32 A frag / 4x16 B frag (2 VGPRs)
typedef __attribute__((ext_vector_type(8))) float v8f; // 16x16 f32 C/D frag (8 VGPRs)

#define N_TOT            16384      // B*L = 32*512
#define TILES            1024       // N_TOT / 16
#define WAVES_PER_BLOCK  8          // 256 threads = 8 wave32
#define IBLOCKS          (TILES / WAVES_PER_BLOCK)   // 128
#define JSPLIT           8
#define JT_PER_BLOCK     (TILES / JSPLIT)            // 128 j-tiles per block
#define FLT_BIG          3.402823466e38f

__global__ __launch_bounds__(256)
void auc_pair_kernel(const int* __restrict__ yt, const float* __restrict__ yp,
                     float* __restrict__ partials)
{
    const int lane   = threadIdx.x & 31;
    const int wave   = threadIdx.x >> 5;
    const int lane15 = lane & 15;
    const bool lo    = lane < 16;           // lanes 0-15 carry the K=0 slot

    // ---- A fragments for this wave's i-tile (loaded once) ----
    // Documented A 16x4 f32 layout: lanes 0-15 / VGPR0 hold K=0 (M = lane).
    const int itile = blockIdx.x * WAVES_PER_BLOCK + wave;
    const int mi    = itile * 16 + lane15;
    const int   ti  = yt[mi];
    const float si  = yp[mi];
    const float pos = (ti == 1) ? 1.0f : 0.0f;

    v2f aPA; aPA.x = lo ? pos * (1.0f - si) : 0.0f; aPA.y = 0.0f;  // A1[i][0] = pos_i * a_i
    v2f aP;  aP.x  = lo ? pos               : 0.0f; aP.y  = 0.0f;  // A2[i][0] = pos_i

    v8f acc = {};   // per-lane hinge accumulator (16x16 C/D layout)

    const int jbase = blockIdx.y * JT_PER_BLOCK * 16;
    for (int jt = 0; jt < JT_PER_BLOCK; ++jt) {
        // ---- B fragments for this j-tile ----
        // B 4x16 f32 layout: lanes 0-15 / VGPR0 hold K=0 (N = lane).
        const int nj    = jbase + jt * 16 + lane15;   // lanes 16-31 duplicate, then masked
        const int   tj  = yt[nj];
        const float sj  = yp[nj];
        const float neg = (tj == 0) ? 1.0f : 0.0f;

        v2f bN;  bN.x  = lo ? neg      : 0.0f; bN.y  = 0.0f;       // B1[0][j] = neg_j
        v2f bNB; bNB.x = lo ? neg * sj : 0.0f; bNB.y = 0.0f;       // B2[0][j] = neg_j * b_j

        // D = (pos*a) x (neg) ; D += (pos) x (neg*b)   => pos_i*neg_j*(a_i + b_j)
        v8f c = {};
        c = __builtin_amdgcn_wmma_f32_16x16x4_f32(false, aPA, false, bN,
                                                  (short)0, c, false, false);
        c = __builtin_amdgcn_wmma_f32_16x16x4_f32(false, aP, false, bNB,
                                                  (short)0, c, false, false);

        // masked hinge = max(0, D) via single v_med3_f32; accumulate
#pragma unroll
        for (int k = 0; k < 8; ++k)
            acc[k] += __builtin_amdgcn_fmed3f(c[k], 0.0f, FLT_BIG);
    }

    // ---- deterministic reduction: lane -> wave -> block ----
    float sum = 0.0f;
#pragma unroll
    for (int k = 0; k < 8; ++k) sum += acc[k];

    for (int off = 16; off > 0; off >>= 1)
        sum += __shfl_down(sum, off, 32);

    __shared__ float lsum[WAVES_PER_BLOCK];
    if (lane == 0) lsum[wave] = sum;
    __syncthreads();

    if (threadIdx.x == 0) {
        float bsum = 0.0f;
        for (int w = 0; w < WAVES_PER_BLOCK; ++w) bsum += lsum[w];   // fixed order
        partials[blockIdx.y * IBLOCKS + blockIdx.x] = bsum;
    }
}

__global__ __launch_bounds__(256)
void auc_final_kernel(const int* __restrict__ yt, const float* __restrict__ partials,
                      float* __restrict__ out)
{
    __shared__ float fs[256];
    __shared__ int   is[256];
    const int t = threadIdx.x;

    float s = 0.0f;
    for (int i = t; i < TILES; i += 256) s += partials[i];   // fixed strided order
    int c = 0;
    for (int i = t; i < N_TOT; i += 256) c += (yt[i] == 1);

    fs[t] = s; is[t] = c;
    __syncthreads();
    for (int stride = 128; stride > 0; stride >>= 1) {       // fixed-order tree
        if (t < stride) { fs[t] += fs[t + stride]; is[t] += is[t + stride]; }
        __syncthreads();
    }
    if (t == 0) {
        const float npos = (float)is[0];
        const float nneg = (float)(N_TOT - is[0]);
        out[0] = fs[0] / (npos * nneg);
    }
}

extern "C" void kernel_launch(void* const* d_in, const int* in_sizes, int n_in,
                              void* d_out, int out_size, void* d_ws, size_t ws_size,
                              hipStream_t stream)
{
    const int*   yt = (const int*)d_in[0];    // y_true, int32, N_TOT
    const float* yp = (const float*)d_in[1];  // y_pred, float32, N_TOT
    float* partials = (float*)d_ws;           // TILES floats (4 KB)
    (void)in_sizes; (void)n_in; (void)out_size; (void)ws_size;

    dim3 grid(IBLOCKS, JSPLIT);
    auc_pair_kernel<<<grid, 256, 0, stream>>>(yt, yp, partials);
    auc_final_kernel<<<1, 256, 0, stream>>>(yt, partials, (float*)d_out);
}